// SpatialPredicate_6236292514515
// MI455X (gfx1250) — compile-verified
//
#include <hip/hip_runtime.h>

// close_to predicate: eps - relu(smooth signed distance), B=8 T=512 V1=64 V2=256 F=128
// 1 workgroup (256 thr = 8 waves32) per (b,t). pn = verts1 x normals^T is a
// [64x3]x[3x128] GEMM -> V_WMMA_F32_16X16X4_F32 (K padded 3->4).
// Operands staged in LDS as 4 coordinate planes {x,y,z,0} so each lane's A/B
// operand pair is two unconditional ds_loads from one cndmask'd base
// (lo lanes: planes {0,1}; hi lanes: planes {2,3}) -- no EXEC divergence.
// Smooth max/min via stabilized logsumexp with associative (m,s) partial merge.

typedef __attribute__((ext_vector_type(2))) float v2f;
typedef __attribute__((ext_vector_type(8))) float v8f;

#define TAU_F     0.01f
#define INV_TAU_F 100.0f
#define EPS_F     1.0f

#define DIM_B  8
#define DIM_T  512
#define DIM_V1 64
#define DIM_V2 256
#define DIM_F  128
#define PITCH  130   // 128 + 2 pad: avoids 64-way LDS bank conflicts in reduction

__launch_bounds__(256)
__global__ void close_to_kernel(const float* __restrict__ verts1,
                                const float* __restrict__ verts2,
                                const long long* __restrict__ faces,
                                const float* __restrict__ normals,
                                float* __restrict__ out)
{
    __shared__ float s_plane[DIM_V1 * PITCH];  // plane_d[v][f]
    __shared__ float s_v1[4 * DIM_V1];         // verts1 planes: x,y,z,zero
    __shared__ float s_nrm[4 * DIM_F];         // normal planes: x,y,z,zero
    __shared__ float s_cn[DIM_F];              // <centroid_f, n_f>
    __shared__ float s_pm[DIM_V1 * 4];         // partial max  (face logsumexp)
    __shared__ float s_ps[DIM_V1 * 4];         // partial sum
    __shared__ float s_sd[DIM_V1];             // smooth-max over faces per vertex

    const int bt  = blockIdx.x;                // b*T + t
    const int tid = threadIdx.x;
    const float* __restrict__ v1 = verts1 + (size_t)bt * DIM_V1 * 3;
    const float* __restrict__ v2 = verts2 + (size_t)bt * DIM_V2 * 3;

    // ---------- Phase 0+1: stage planar operands (+zero planes); cn[f] ----------
    if (tid < 3 * DIM_V1) {                    // 192 threads: verts1 -> planar LDS
        const int v = tid / 3, c = tid % 3;
        s_v1[c * DIM_V1 + v] = v1[tid];
    }
    if (tid < DIM_V1)  s_v1[3 * DIM_V1 + tid] = 0.0f;     // K=3 zero pad plane
    for (int i = tid; i < 3 * DIM_F; i += 256) {          // normals -> planar LDS
        const int f = i / 3, c = i % 3;
        s_nrm[c * DIM_F + f] = normals[i];
    }
    if (tid < DIM_F)   s_nrm[3 * DIM_F + tid] = 0.0f;     // K=3 zero pad plane
    if (tid < DIM_F) {                         // cn[f] from global gathers
        const int f = tid;
        const long long i0 = faces[f * 3 + 0];
        const long long i1 = faces[f * 3 + 1];
        const long long i2 = faces[f * 3 + 2];
        const float nx = normals[f * 3 + 0];
        const float ny = normals[f * 3 + 1];
        const float nz = normals[f * 3 + 2];
        const float third = 1.0f / 3.0f;
        const float cx = (v2[i0 * 3 + 0] + v2[i1 * 3 + 0] + v2[i2 * 3 + 0]) * third;
        const float cy = (v2[i0 * 3 + 1] + v2[i1 * 3 + 1] + v2[i2 * 3 + 1]) * third;
        const float cz = (v2[i0 * 3 + 2] + v2[i1 * 3 + 2] + v2[i2 * 3 + 2]) * third;
        s_cn[f] = cx * nx + cy * ny + cz * nz;
    }
    __syncthreads();

    // ---------------- Phase 2: pn tiles via WMMA f32 16x16x4 ----------------
    // Wave w owns faces [16w,16w+16); loops over 4 vertex tiles of 16.
    {
        const int wave  = tid >> 5;            // 0..7  (N tile)
        const int lane  = tid & 31;
        const int lhalf = lane & 15;
        const bool hi   = lane >= 16;
        const int pbase = hi ? 2 : 0;          // plane pair: {x,y} lo / {z,0} hi

        // B (4x16 = normals^T): lane = face column; lanes 0-15 carry K={0,1}
        // in the 2 VGPRs, lanes 16-31 carry K={2,3} (K=3 row is the zero plane).
        const int f = wave * 16 + lhalf;
        const float* bp = &s_nrm[pbase * DIM_F + f];
        v2f bmat;
        bmat.x = bp[0];                        // plane pbase
        bmat.y = bp[DIM_F];                    // plane pbase+1
        const float cn_f = s_cn[f];

        #pragma unroll
        for (int m = 0; m < 4; ++m) {
            const int vrow = m * 16 + lhalf;   // A row for this lane
            const float* ap = &s_v1[pbase * DIM_V1 + vrow];
            v2f amat;
            amat.x = ap[0];
            amat.y = ap[DIM_V1];

            v8f acc = {};
            // D = A x B + 0 (neg_a, A, neg_b, B, c_mod, C, reuse_a, reuse_b)
            acc = __builtin_amdgcn_wmma_f32_16x16x4_f32(
                false, amat, false, bmat, (short)0, acc, false, false);

            // D layout: lane -> column f; VGPR j -> row (j + hi*8) of tile m.
            const int vbase = m * 16 + (hi ? 8 : 0);
            #pragma unroll
            for (int j = 0; j < 8; ++j) {
                s_plane[(vbase + j) * PITCH + f] = acc[j] - cn_f;
            }
        }
    }
    __syncthreads();

    // ------- Phase 3a: partial logsumexp over faces (256 thr, 32 faces each) -------
    {
        const int v = tid >> 2;                // vertex 0..63
        const int q = tid & 3;                 // face quarter
        const float* row = &s_plane[v * PITCH + q * 32];
        float m = -INFINITY;
        #pragma unroll 4
        for (int fi = 0; fi < 32; ++fi) m = fmaxf(m, row[fi]);
        float s = 0.0f;
        #pragma unroll 4
        for (int fi = 0; fi < 32; ++fi) s += __expf((row[fi] - m) * INV_TAU_F);
        s_pm[v * 4 + q] = m;
        s_ps[v * 4 + q] = s;
    }
    __syncthreads();

    // ------- Phase 3b: merge partials -> sd_pts[v] = tau*logsumexp(x/tau) -------
    if (tid < DIM_V1) {
        const int v = tid;
        float m = fmaxf(fmaxf(s_pm[v * 4 + 0], s_pm[v * 4 + 1]),
                        fmaxf(s_pm[v * 4 + 2], s_pm[v * 4 + 3]));
        float s = 0.0f;
        #pragma unroll
        for (int q = 0; q < 4; ++q)
            s += s_ps[v * 4 + q] * __expf((s_pm[v * 4 + q] - m) * INV_TAU_F);
        s_sd[v] = m + TAU_F * __logf(s);
    }
    __syncthreads();

    // ------- Phase 4: smooth min over 64 vertices (wave-0 butterfly) -------
    if (tid < 32) {
        const float x0 = s_sd[tid];
        const float x1 = s_sd[tid + 32];
        float m = fminf(x0, x1);               // per-lane partial (min version)
        float s = __expf((m - x0) * INV_TAU_F) + __expf((m - x1) * INV_TAU_F);
        #pragma unroll
        for (int off = 16; off > 0; off >>= 1) {
            const float om = __shfl_xor(m, off, 32);
            const float os = __shfl_xor(s, off, 32);
            const float nm = fminf(m, om);
            s = s * __expf((nm - m) * INV_TAU_F) + os * __expf((nm - om) * INV_TAU_F);
            m = nm;
        }
        if (tid == 0) {
            const float sd = m - TAU_F * __logf(s);   // -tau*logsumexp(-x/tau)
            out[bt] = EPS_F - fmaxf(sd, 0.0f);        // eps - relu(sd)
        }
    }
}

extern "C" void kernel_launch(void* const* d_in, const int* in_sizes, int n_in,
                              void* d_out, int out_size, void* d_ws, size_t ws_size,
                              hipStream_t stream) {
    const float*     verts1  = (const float*)d_in[0];       // [B,T,V1,3] f32
    const float*     verts2  = (const float*)d_in[1];       // [B,T,V2,3] f32
    const long long* faces   = (const long long*)d_in[2];   // [F,3] int64
    const float*     normals = (const float*)d_in[3];       // [F,3] f32
    float* out = (float*)d_out;                             // [B,T,1] f32

    (void)in_sizes; (void)n_in; (void)out_size; (void)d_ws; (void)ws_size;

    close_to_kernel<<<DIM_B * DIM_T, 256, 0, stream>>>(verts1, verts2, faces, normals, out);
}